// PostProcessor_61976378081525
// MI455X (gfx1250) — compile-verified
//
#include <hip/hip_runtime.h>
#include <stdint.h>

#define HH 512
#define WW 1024
#define HWPIX (HH * WW)
#define NCLS 19
#define KEEPK 128
#define CAP 131072
#define CTHRESH 0.1f

typedef __attribute__((ext_vector_type(2))) float v2f;
typedef __attribute__((ext_vector_type(8))) float v8f;

// ---------------------------------------------------------------- init
__global__ void k_init(unsigned int* counter) {
    if (threadIdx.x == 0 && blockIdx.x == 0) *counter = 0u;
}

// ---------------------------------------------------------------- semantic argmax + thing mask
__global__ void k_semantic(const float* __restrict__ logits,
                           const int* __restrict__ thing_ids, int n_thing,
                           float* __restrict__ sem_out, float* __restrict__ thing_out) {
    int p = blockIdx.x * blockDim.x + threadIdx.x;
    if (p >= HWPIX) return;
    const float* l = logits + (size_t)p * NCLS;
    float best = l[0];
    int bi = 0;
#pragma unroll
    for (int c = 1; c < NCLS; ++c) {
        float v = l[c];
        if (v > best) { best = v; bi = c; }   // strict > == first occurrence (jnp.argmax)
    }
    float th = 0.0f;
    for (int i = 0; i < n_thing; ++i)
        if (thing_ids[i] == bi) th = 1.0f;
    sem_out[p]  = (float)bi;
    thing_out[p] = th;
}

// ---------------------------------------------------------------- NMS pass 1: horizontal 7-tap max of thresholded heatmap
__device__ __forceinline__ float thresh(float v) { return v > CTHRESH ? v : 0.0f; }

__global__ void k_rowmax(const float* __restrict__ heat, float* __restrict__ rowmax) {
    int p = blockIdx.x * blockDim.x + threadIdx.x;
    if (p >= HWPIX) return;
    int y = p / WW, x = p % WW;
    float m = 0.0f;  // zero padding valid: thresholded values >= 0
#pragma unroll
    for (int dx = -3; dx <= 3; ++dx) {
        int xx = x + dx;
        if (xx >= 0 && xx < WW) {
            float v = thresh(heat[y * WW + xx]);
            m = v > m ? v : m;
        }
    }
    rowmax[p] = m;
}

// ---------------------------------------------------------------- NMS pass 2: vertical 7-tap max, peak test, candidate compaction
__global__ void k_nms(const float* __restrict__ heat, const float* __restrict__ rowmax,
                      float* __restrict__ out_hm,
                      unsigned long long* __restrict__ cand, unsigned int* __restrict__ counter) {
    int p = blockIdx.x * blockDim.x + threadIdx.x;
    if (p >= HWPIX) return;
    int y = p / WW, x = p % WW;
    float pooled = 0.0f;
#pragma unroll
    for (int dy = -3; dy <= 3; ++dy) {
        int yy = y + dy;
        if (yy >= 0 && yy < HH) {
            float v = rowmax[yy * WW + x];
            pooled = v > pooled ? v : pooled;
        }
    }
    float t  = thresh(heat[p]);
    float hm = (pooled == t) ? t : 0.0f;
    out_hm[p] = hm;
    if (hm > 0.0f) {
        // key: score bits (monotone for >=0 floats) | ~idx (smaller idx wins ties, like lax.top_k)
        unsigned long long key =
            ((unsigned long long)__float_as_uint(hm) << 32) | (unsigned int)(~(unsigned int)p);
        unsigned int pos = atomicAdd(counter, 1u);
        if (pos < CAP) cand[pos] = key;
    }
}

// ---------------------------------------------------------------- top-K selection + WMMA A-matrix packing
__global__ void k_topk(const unsigned long long* __restrict__ cand,
                       const unsigned int* __restrict__ counter,
                       float* __restrict__ scores_ws,
                       float* __restrict__ A0, float* __restrict__ A1) {
    __shared__ unsigned long long smem[1024];
    __shared__ unsigned long long prevKey;
    __shared__ float s_scores[KEEPK], s_cy[KEEPK], s_cx[KEEPK];

    int tid = threadIdx.x;
    unsigned int cnt = *counter;
    int M = (cnt < CAP) ? (int)cnt : CAP;
    if (tid == 0) prevKey = ~0ull;
    __syncthreads();

    for (int r = 0; r < KEEPK; ++r) {
        unsigned long long pk = prevKey;
        unsigned long long local = 0ull;
        for (int i = tid; i < M; i += 1024) {
            unsigned long long k = cand[i];
            if (k < pk && k > local) local = k;   // keys unique: strict-below-prev selects next
        }
        smem[tid] = local;
        __syncthreads();
        for (int s = 512; s > 0; s >>= 1) {
            if (tid < s && smem[tid + s] > smem[tid]) smem[tid] = smem[tid + s];
            __syncthreads();
        }
        if (tid == 0) {
            unsigned long long k = smem[0];
            float sc = 0.0f;
            unsigned int p = 0u;
            if (k != 0ull) {
                sc = __uint_as_float((unsigned int)(k >> 32));
                p  = ~((unsigned int)(k & 0xffffffffull));
            }
            s_scores[r] = sc;
            s_cy[r] = (float)(p / WW);
            s_cx[r] = (float)(p % WW);
            prevKey = k;
        }
        __syncthreads();
    }

    if (tid < KEEPK) scores_ws[tid] = s_scores[tid];

    // Pack A tiles in the 16x4 f32 WMMA lane layout:
    // lane<16:  v0 = A[row][K=0] = cy, v1 = A[row][K=1] = cx      (row = t*16+lane)
    // lane>=16: v0 = A[row][K=2] = |c|^2 (or 1e30 if invalid), v1 = A[row][K=3] = 0
    for (int i = tid; i < 256; i += 1024) {
        int l = i & 31, t = i >> 5;
        float a0, a1;
        if (l < 16) {
            int row = t * 16 + l;
            a0 = s_cy[row];
            a1 = s_cx[row];
        } else {
            int row = t * 16 + (l - 16);
            float s = s_scores[row];
            a0 = (s > 0.0f) ? (s_cy[row] * s_cy[row] + s_cx[row] * s_cx[row]) : 1e30f;
            a1 = 0.0f;
        }
        A0[i] = a0;
        A1[i] = a1;
    }
}

// ---------------------------------------------------------------- WMMA distance argmin + fused outputs
// D = A(16 centers x 4) * B(4 x 16 pixels): value = |c|^2 - 2 c . p  (per-pixel |p|^2 dropped; constant in argmin)
__global__ void k_assign(const float* __restrict__ offsets,
                         const float* __restrict__ A0, const float* __restrict__ A1,
                         const float* __restrict__ scores_ws, const float* __restrict__ thing_ws,
                         float* __restrict__ out_inst, float* __restrict__ out_score) {
    int lane = threadIdx.x & 31;
    int wave = threadIdx.x >> 5;                 // 8 waves / block
    int p = (blockIdx.x * 8 + wave) * 16 + (lane & 15);
    bool hi = lane >= 16;

    float y = (float)(p / WW);
    float x = (float)(p % WW);
    float offy = offsets[2 * p + 0];
    float offx = offsets[2 * p + 1];

    // B column per pixel: [-2*py, -2*px, 1, 0]; lanes 16-31 hold K=2,3 rows
    v2f b;
    b.x = hi ? 1.0f : -2.0f * (y + offy);
    b.y = hi ? 0.0f : -2.0f * (x + offx);

    float bestV = 3.4028235e38f;
    int   bestI = 0;

#pragma unroll
    for (int t = 0; t < 8; ++t) {
        v2f a;
        a.x = A0[t * 32 + lane];
        a.y = A1[t * 32 + lane];
        v8f c = {};
        v8f d = __builtin_amdgcn_wmma_f32_16x16x4_f32(
            /*neg_a=*/false, a, /*neg_b=*/false, b,
            /*c_mod=*/(short)0, c, /*reuse_a=*/false, /*reuse_b=*/false);
        int baseIdx = t * 16 + (hi ? 8 : 0);
#pragma unroll
        for (int v = 0; v < 8; ++v) {
            float val = d[v];
            int idx = baseIdx + v;                // increasing per lane -> strict < keeps smallest idx
            if (val < bestV) { bestV = val; bestI = idx; }
        }
    }

    // combine with partner lane (other half of the center dimension); lexicographic tie-break
    float pv = __shfl_xor(bestV, 16, 32);
    int   pi = __shfl_xor(bestI, 16, 32);
    if (pv < bestV || (pv == bestV && pi < bestI)) { bestV = pv; bestI = pi; }

    if (lane < 16) {
        float th = thing_ws[p];
        out_inst[p]  = (float)(bestI + 1) * th;
        out_score[p] = scores_ws[bestI] * th;
    }
}

// ---------------------------------------------------------------- launcher
extern "C" void kernel_launch(void* const* d_in, const int* in_sizes, int n_in,
                              void* d_out, int out_size, void* d_ws, size_t ws_size,
                              hipStream_t stream) {
    const float* logits = (const float*)d_in[0];   // [1,H,W,19]
    const float* heat   = (const float*)d_in[1];   // [H,W,1]
    const float* offs   = (const float*)d_in[2];   // [H,W,2]
    const int*   tids   = (const int*)d_in[3];     // [8]
    int n_thing = in_sizes[3];

    float* out = (float*)d_out;
    float* out_inst  = out;                 // instance_seg
    float* out_hm    = out + HWPIX;         // hm
    float* out_score = out + 2 * HWPIX;     // score_map
    float* out_sem   = out + 3 * HWPIX;     // semantic

    char* ws = (char*)d_ws;
    float*              thing_ws  = (float*)ws;                                  // HW floats
    float*              row_ws    = (float*)(ws + (size_t)HWPIX * 4);            // HW floats
    unsigned long long* cand      = (unsigned long long*)(ws + (size_t)2 * HWPIX * 4); // CAP u64
    unsigned int*       counter   = (unsigned int*)(ws + (size_t)2 * HWPIX * 4 + (size_t)CAP * 8);
    float*              scores_ws = (float*)(ws + (size_t)2 * HWPIX * 4 + (size_t)CAP * 8 + 64);
    float*              A0        = scores_ws + KEEPK;      // 256 floats
    float*              A1        = A0 + 256;               // 256 floats

    k_init<<<1, 1, 0, stream>>>(counter);
    k_semantic<<<HWPIX / 256, 256, 0, stream>>>(logits, tids, n_thing, out_sem, thing_ws);
    k_rowmax<<<HWPIX / 256, 256, 0, stream>>>(heat, row_ws);
    k_nms<<<HWPIX / 256, 256, 0, stream>>>(heat, row_ws, out_hm, cand, counter);
    k_topk<<<1, 1024, 0, stream>>>(cand, counter, scores_ws, A0, A1);
    k_assign<<<HWPIX / 128, 256, 0, stream>>>(offs, A0, A1, scores_ws, thing_ws, out_inst, out_score);
}